// BertHungarianLoss_62070867362602
// MI455X (gfx1250) — compile-verified
//
#include <hip/hip_runtime.h>
#include <math.h>

#define MM 8            // masked tokens (rows)
#define NBLK 32         // partial-reduction blocks per row

typedef __attribute__((ext_vector_type(2))) float v2f;
typedef __attribute__((ext_vector_type(8))) float v8f;

// Online softmax merge: (m,s) <- merge((m,s),(m2,s2)) with s = sum exp(x - m)
__device__ __forceinline__ void smerge(float& m, float& s, float m2, float s2) {
    float nm = fmaxf(m, m2);
    if (nm == -INFINITY) { m = nm; s = 0.0f; return; }
    s = s * __expf(m - nm) + s2 * __expf(m2 - nm);
    m = nm;
}

// Kernel 1: per-row (max, sum-exp) partials. Grid: (NBLK, MM). Block: 256.
__global__ void bhl_softmax_partials(const float* __restrict__ logits,
                                     float* __restrict__ pm, float* __restrict__ ps,
                                     int V) {
    const int row = blockIdx.y;
    const float* x = logits + (long long)row * V;
    const int t = threadIdx.x;
    const int stride = gridDim.x * blockDim.x;
    float m = -INFINITY, s = 0.0f;
    for (int idx = blockIdx.x * blockDim.x + t; idx < V; idx += stride) {
        float v  = x[idx];
        float nm = fmaxf(m, v);
        s = s * __expf(m - nm) + __expf(v - nm);
        m = nm;
    }
    __shared__ float sm[256], ss[256];
    sm[t] = m; ss[t] = s;
    __syncthreads();
    for (int off = 128; off > 0; off >>= 1) {
        if (t < off) {
            float m1 = sm[t], s1 = ss[t];
            smerge(m1, s1, sm[t + off], ss[t + off]);
            sm[t] = m1; ss[t] = s1;
        }
        __syncthreads();
    }
    if (t == 0) {
        pm[row * gridDim.x + blockIdx.x] = sm[0];
        ps[row * gridDim.x + blockIdx.x] = ss[0];
    }
}

// Kernel 2 (single block, 8 wave32 waves):
//   scores = Onehot[P,64] x vec(Q)[64] via chains of v_wmma_f32_16x16x4_f32,
//   first-max argmax, then CE loss on the best assignment.
__global__ void __launch_bounds__(256) bhl_assign_score(
        const float* __restrict__ logits, const int* __restrict__ target,
        const int* __restrict__ perms, const float* __restrict__ pm,
        const float* __restrict__ ps, int V, int P, int nblk,
        float* __restrict__ out) {
    __shared__ float sh_max[MM], sh_sum[MM];
    __shared__ float sh_q[64];   // Q[j][i] = softmax(logits)[j, target[i]], flat j*8+i
    __shared__ float sh_z[64];   // z[j][i] = logits[j, target[i]]
    __shared__ float rs[256];
    __shared__ int   ri[256];
    const int t = threadIdx.x;

    // Phase A: fold the NBLK partials per row.
    if (t < MM) {
        float m = -INFINITY, s = 0.0f;
        for (int b = 0; b < nblk; ++b)
            smerge(m, s, pm[t * nblk + b], ps[t * nblk + b]);
        sh_max[t] = m; sh_sum[t] = s;
    }
    __syncthreads();

    // Phase B: gather the 64 relevant logits, build Q.
    if (t < 64) {
        int j = t >> 3, i = t & 7;
        float z = logits[(long long)j * V + target[i]];
        sh_z[t] = z;
        sh_q[t] = __expf(z - sh_max[j]) / sh_sum[j];
    }
    __syncthreads();

    // Phase C: WMMA permutation scoring. 16 perms per chain, K=64 in 16 steps of 4.
    // A-fragment (16x4 f32): lane l -> row M=l&15; VGPR0/1 hold K-offsets {0,1}
    // (lanes 0-15) or {2,3} (lanes 16-31). B broadcast across N so every D column
    // holds the score; D: acc[r] on any lane = score[p_base + 8*(lane>>4) + r].
    const int lane = t & 31;
    const int wave = t >> 5;            // 8 waves, wave-uniform loop bounds (EXEC all 1s)
    const int half = lane >> 4;
    const int m16  = lane & 15;
    const int chains = P >> 4;          // P = 40320 -> 2520 chains
    float best_s = -INFINITY;
    int   best_i = 0x7fffffff;

    for (int it = wave; it < chains; it += 8) {
        const int p_base = it << 4;
        const int p = p_base + m16;     // this lane's A row (both K-halves load same perm)
        const int4* prow = (const int4*)perms + (long long)p * 2;
        int4 pa = prow[0];
        int4 pb = prow[1];
        int pv[8] = { pa.x, pa.y, pa.z, pa.w, pb.x, pb.y, pb.z, pb.w };
        v8f acc = {0.f, 0.f, 0.f, 0.f, 0.f, 0.f, 0.f, 0.f};
#pragma unroll
        for (int c = 0; c < 16; ++c) {
            const int j  = c >> 1;              // K index 4c..4c+3 stays within token j
            const int bi = (c & 1) * 4;
            const int i0 = bi + (half << 1);    // this lane's two K slots
            v2f a, b;
            a.x = (pv[j] == i0)     ? 1.0f : 0.0f;
            a.y = (pv[j] == i0 + 1) ? 1.0f : 0.0f;
            b.x = sh_q[4 * c + (half << 1)];
            b.y = sh_q[4 * c + (half << 1) + 1];
            acc = __builtin_amdgcn_wmma_f32_16x16x4_f32(
                      false, a, false, b, (short)0, acc, false, false);
        }
#pragma unroll
        for (int r = 0; r < 8; ++r) {
            int pidx = p_base + (half << 3) + r;
            float sc = acc[r];
            if (sc > best_s || (sc == best_s && pidx < best_i)) {
                best_s = sc; best_i = pidx;
            }
        }
    }

    // Block argmax with first-index tie-break (matches jnp.argmax).
    rs[t] = best_s; ri[t] = best_i;
    __syncthreads();
    for (int off = 128; off > 0; off >>= 1) {
        if (t < off) {
            float s2 = rs[t + off]; int i2 = ri[t + off];
            if (s2 > rs[t] || (s2 == rs[t] && i2 < ri[t])) { rs[t] = s2; ri[t] = i2; }
        }
        __syncthreads();
    }

    // Phase D: emit loss[8] then target_best[8] (as float).
    if (t == 0) {
        int best = ri[0];
        for (int j = 0; j < MM; ++j) {
            int i = perms[best * MM + j];
            float loss = -(sh_z[j * 8 + i] - sh_max[j] - __logf(sh_sum[j]));
            out[j]      = loss;
            out[MM + j] = (float)target[i];
        }
    }
}

extern "C" void kernel_launch(void* const* d_in, const int* in_sizes, int n_in,
                              void* d_out, int out_size, void* d_ws, size_t ws_size,
                              hipStream_t stream) {
    const float* logits = (const float*)d_in[0];   // [8, V] f32
    const int*   target = (const int*)d_in[1];     // [8] i32
    const int*   perms  = (const int*)d_in[2];     // [P, 8] i32
    float* out = (float*)d_out;                    // [8] loss + [8] target_best

    const int V = in_sizes[0] / MM;
    const int P = in_sizes[2] / MM;

    float* pm = (float*)d_ws;            // [MM * NBLK]
    float* ps = pm + MM * NBLK;          // [MM * NBLK]

    dim3 g1(NBLK, MM);
    bhl_softmax_partials<<<g1, 256, 0, stream>>>(logits, pm, ps, V);
    bhl_assign_score<<<1, 256, 0, stream>>>(logits, target, perms, pm, ps, V, P, NBLK, out);
}